// SGC_86964497809763
// MI455X (gfx1250) — compile-verified
//
#include <hip/hip_runtime.h>
#include <hip/hip_bf16.h>

// ---------------------------------------------------------------------------
// SGC (2-hop simplified graph conv) for MI455X / gfx1250.
//   deg/norm  -> fp32 atomics (resolved in L2; node data fits in 192MB L2)
//   2x hop    -> gather float4 rows + global_atomic_add_f32 scatter (BW bound)
//   linear    -> V_WMMA_F32_16X16X4_F32, weight staged in LDS
// ---------------------------------------------------------------------------

#define N_NODES 100000
#define N_EDGES 1600000
#define DFEAT   64
#define NTILES  (N_NODES / 16)   // 6250 exact

typedef __attribute__((ext_vector_type(2))) float v2f;
typedef __attribute__((ext_vector_type(8))) float v8f;

// ---- zero fill (float4 granularity) ---------------------------------------
__global__ void sgc_zero_f4(float4* __restrict__ p, int n4) {
    int t = blockIdx.x * blockDim.x + threadIdx.x;
    if (t < n4) p[t] = make_float4(0.f, 0.f, 0.f, 0.f);
}

// ---- in-degree ------------------------------------------------------------
__global__ void sgc_degree(const long long* __restrict__ dst,
                           float* __restrict__ deg, int nEdges) {
    int e = blockIdx.x * blockDim.x + threadIdx.x;
    if (e < nEdges) atomicAdd(&deg[dst[e]], 1.0f);
}

// ---- norm = rsqrt(max(deg,1)) in place ------------------------------------
__global__ void sgc_norm(float* __restrict__ deg, int nNodes) {
    int i = blockIdx.x * blockDim.x + threadIdx.x;
    if (i < nNodes) deg[i] = rsqrtf(fmaxf(deg[i], 1.0f));
}

// ---- edge scatter: agg[dst] += h[src] * norm[src] -------------------------
// 16 threads per edge, one float4 slice each.
__global__ void sgc_scatter(const float* __restrict__ h,
                            const float* __restrict__ norm,
                            const long long* __restrict__ src,
                            const long long* __restrict__ dst,
                            float* __restrict__ agg, int nEdges) {
    long long t = (long long)blockIdx.x * blockDim.x + threadIdx.x;
    int e    = (int)(t >> 4);
    int part = (int)(t & 15);
    if (e >= nEdges) return;
    long long s = src[e];
    long long d = dst[e];
    float ns = norm[s];
    const float4 v = *(const float4*)(h + s * DFEAT + part * 4);
    float* o = agg + d * DFEAT + part * 4;
    atomicAdd(o + 0, v.x * ns);
    atomicAdd(o + 1, v.y * ns);
    atomicAdd(o + 2, v.z * ns);
    atomicAdd(o + 3, v.w * ns);
}

// ---- post-scale: h[i] *= norm[i] (float4 per thread) ----------------------
__global__ void sgc_scale(float* __restrict__ h,
                          const float* __restrict__ norm, int nNodes) {
    int t = blockIdx.x * blockDim.x + threadIdx.x;   // nNodes*16 threads
    if (t >= nNodes * 16) return;
    int node = t >> 4;
    float nv = norm[node];
    float4* p = (float4*)h + t;
    float4 v = *p;
    v.x *= nv; v.y *= nv; v.z *= nv; v.w *= nv;
    *p = v;
}

// ---- WMMA GEMM: out = h @ W^T + bias --------------------------------------
// One wave per 16-row strip; 4 column tiles of 16; K=64 in 16 steps of 4.
// A 16x4 f32 layout: lanes 0-15 hold {K=k,k+1}, lanes 16-31 hold {K=k+2,k+3}.
// B 4x16 f32 layout mirrors A: lanes 0-15 = rows K=k,k+1 at N=lane,
//                              lanes 16-31 = rows K=k+2,k+3 at N=lane-16.
// D 16x16 f32: VGPR j -> M=j (lanes 0-15), M=8+j (lanes 16-31), N=lane&15.
__global__ void __launch_bounds__(256)
sgc_gemm_wmma(const float* __restrict__ h, const float* __restrict__ w,
              const float* __restrict__ bias, float* __restrict__ out,
              int nTiles) {
    __shared__ float lw[DFEAT * DFEAT];   // 16 KB: W row-major, W[o][k]

    // cooperative load of the 64x64 weight into LDS
    {
        const float4* wsrc = (const float4*)w;
        float4* ldst = (float4*)lw;
        #pragma unroll
        for (int i = 0; i < 4; ++i)
            ldst[threadIdx.x + i * 256] = wsrc[threadIdx.x + i * 256];
    }
    __syncthreads();

    const int wave = threadIdx.x >> 5;
    const int lane = threadIdx.x & 31;
    const int tile = blockIdx.x * 8 + wave;

    if (tile < nTiles) {              // wave-uniform: EXEC stays all-ones
        const int m0  = tile * 16;
        const int hi  = lane >> 4;    // 0: K pair {k,k+1}; 1: {k+2,k+3}
        const int l15 = lane & 15;
        const float* hrow = h + (long long)(m0 + l15) * DFEAT;

        v8f acc[4];
        #pragma unroll
        for (int n = 0; n < 4; ++n) acc[n] = (v8f){0.f,0.f,0.f,0.f,0.f,0.f,0.f,0.f};

        #pragma unroll
        for (int k = 0; k < DFEAT; k += 4) {
            const int kk = k + hi * 2;
            v2f a = *(const v2f*)(hrow + kk);
            #pragma unroll
            for (int n = 0; n < 4; ++n) {
                v2f b = *(const v2f*)(lw + (n * 16 + l15) * DFEAT + kk);
                acc[n] = __builtin_amdgcn_wmma_f32_16x16x4_f32(
                    /*neg_a=*/false, a, /*neg_b=*/false, b,
                    /*c_mod=*/(short)0, acc[n],
                    /*reuse_a=*/false, /*reuse_b=*/false);
            }
        }

        const int rbase = m0 + hi * 8;
        #pragma unroll
        for (int n = 0; n < 4; ++n) {
            const int col = n * 16 + l15;
            const float bv = bias[col];
            #pragma unroll
            for (int j = 0; j < 8; ++j)
                out[(long long)(rbase + j) * DFEAT + col] = acc[n][j] + bv;
        }
    }
}

// ---------------------------------------------------------------------------
extern "C" void kernel_launch(void* const* d_in, const int* in_sizes, int n_in,
                              void* d_out, int out_size, void* d_ws, size_t ws_size,
                              hipStream_t stream) {
    const float*     feat   = (const float*)d_in[0];
    const long long* src    = (const long long*)d_in[1];   // jnp.int64
    const long long* dst    = (const long long*)d_in[2];   // jnp.int64
    const float*     weight = (const float*)d_in[3];
    const float*     bias   = (const float*)d_in[4];
    float*           out    = (float*)d_out;

    // workspace layout (floats): norm | hA | hB  = 100000 + 2*6.4M ≈ 51.6 MB
    float* ws   = (float*)d_ws;
    float* norm = ws;
    float* hA   = ws + N_NODES;
    float* hB   = ws + N_NODES + (size_t)N_NODES * DFEAT;

    const int totalF  = N_NODES + 2 * N_NODES * DFEAT;   // 12,900,000 (mult of 4)
    const int n4      = totalF / 4;
    sgc_zero_f4<<<(n4 + 255) / 256, 256, 0, stream>>>((float4*)ws, n4);

    sgc_degree<<<(N_EDGES + 255) / 256, 256, 0, stream>>>(dst, norm, N_EDGES);
    sgc_norm<<<(N_NODES + 255) / 256, 256, 0, stream>>>(norm, N_NODES);

    const long long edgeThreads = (long long)N_EDGES * 16;       // 25.6M
    const int edgeBlocks = (int)((edgeThreads + 255) / 256);
    const int nodeThreads = N_NODES * 16;
    const int nodeBlocks  = (nodeThreads + 255) / 256;

    // hop 1: hA = norm ⊙ scatter(feat ⊙ norm)
    sgc_scatter<<<edgeBlocks, 256, 0, stream>>>(feat, norm, src, dst, hA, N_EDGES);
    sgc_scale<<<nodeBlocks, 256, 0, stream>>>(hA, norm, N_NODES);

    // hop 2: hB = norm ⊙ scatter(hA ⊙ norm)
    sgc_scatter<<<edgeBlocks, 256, 0, stream>>>(hA, norm, src, dst, hB, N_EDGES);
    sgc_scale<<<nodeBlocks, 256, 0, stream>>>(hB, norm, N_NODES);

    // linear layer via f32 WMMA
    sgc_gemm_wmma<<<(NTILES + 7) / 8, 256, 0, stream>>>(hB, weight, bias, out, NTILES);
}